// ImgExemplarSelfAttn_17849884082405
// MI455X (gfx1250) — compile-verified
//
#include <hip/hip_runtime.h>
#include <cstdint>
#include <cstddef>

// ---------------------------------------------------------------------------
// Problem constants (match reference)
// ---------------------------------------------------------------------------
#define B_    4
#define C_    256
#define HW_   4096
#define W_    64
#define NEXE  4096          // exemplar column index
#define NPAD  4112          // 257 * 16 (padded N)
#define NT_   257           // n tiles of 16
#define Q_    150
#define HID_  1024
#define CP1   257
#define NEGB  1.0e9f

#define ATP   258           // LDS row stride for 16x256 transposed A tiles (gemm kernels)
#define VSP   264           // LDS row stride for V tile rows (256 data + 8 pad dwords)

typedef float v2f __attribute__((ext_vector_type(2)));
typedef float v8f __attribute__((ext_vector_type(8)));
typedef unsigned v4u __attribute__((ext_vector_type(4)));
typedef int v4i __attribute__((ext_vector_type(4)));
typedef int v8i __attribute__((ext_vector_type(8)));

// Tensor Data Mover availability (device pass only; falls back to coop staging)
#if defined(__HIP_DEVICE_COMPILE__)
#if __has_builtin(__builtin_amdgcn_tensor_load_to_lds) && __has_builtin(__builtin_amdgcn_s_wait_tensorcnt)
#define TDM_OK 1
#endif
#endif

// D = A(16x4) * B(4x16) + C, fp32 WMMA (wave32)
static __device__ __forceinline__ v8f wmma4(v2f a, v2f b, v8f c) {
    return __builtin_amdgcn_wmma_f32_16x16x4_f32(false, a, false, b,
                                                 (short)0, c, false, false);
}

#ifdef TDM_OK
// ---------------------------------------------------------------------------
// Issue one 2D TDM tile load: tile (tx x ty) of a [dim1 x dim0] fp32 tensor
// (row stride s0 elements) -> LDS at byte address lds. D# packing per ISA ch.8.
// w0flags carries data_size/pad fields of D# group1 word0.
// ---------------------------------------------------------------------------
static __device__ __forceinline__ void tdm_load_2d(const float* g, unsigned lds,
                                                   unsigned w0flags,
                                                   unsigned tx, unsigned ty,
                                                   unsigned d0, unsigned d1,
                                                   unsigned s0) {
    const unsigned long long ga = (unsigned long long)(size_t)g;
    v4u g0;
    g0.x = 1u;                                            // count=1, user D#
    g0.y = lds;                                           // lds_addr
    g0.z = (unsigned)(ga & 0xFFFFFFFFull);                // global_addr[31:0]
    g0.w = (unsigned)((ga >> 32) & 0x01FFFFFFull) | (2u << 30);  // addr[56:32] | type=2
    v8i g1;
    g1[0] = (int)w0flags;                                 // wg_mask=0 | data_size | pad cfg
    g1[1] = (int)((d0 & 0xFFFFu) << 16);                  // tensor_dim0[15:0] @ 63:48
    g1[2] = (int)((d0 >> 16) | ((d1 & 0xFFFFu) << 16));   // dim0[31:16] | dim1[15:0]
    g1[3] = (int)((d1 >> 16) | (tx << 16));               // dim1[31:16] | tile_dim0
    g1[4] = (int)ty;                                      // tile_dim1 (tile_dim2=0)
    g1[5] = (int)s0;                                      // tensor_dim0_stride[31:0]
    g1[6] = 0;                                            // stride hi / dim1_stride lo
    g1[7] = 0;
    const v4i z4 = {0, 0, 0, 0};
#if __clang_major__ >= 23
    const v8i z8 = {0, 0, 0, 0, 0, 0, 0, 0};
    __builtin_amdgcn_tensor_load_to_lds(g0, g1, z4, z4, z8, 0);
#else
    __builtin_amdgcn_tensor_load_to_lds(g0, g1, z4, z4, 0);
#endif
}
#endif // TDM_OK

// ---------------------------------------------------------------------------
// 1) exemplar mean over 7x7:   exem[b,c] = mean(exe_feat[b,c,:,:])
// ---------------------------------------------------------------------------
__global__ void k_exe_mean(const float* __restrict__ exe, float* __restrict__ exem) {
    int g = blockIdx.x * blockDim.x + threadIdx.x;          // B_*C_ threads
    if (g >= B_ * C_) return;
    const float* p = exe + (size_t)g * 49;
    float s = 0.f;
    #pragma unroll
    for (int i = 0; i < 49; ++i) s += p[i];
    exem[g] = s * (1.0f / 49.0f);
}

// ---------------------------------------------------------------------------
// 2) build padded concat  Xc[b,c,n] : n<4096 -> img ; n==4096 -> exem ; pad->0
// ---------------------------------------------------------------------------
__global__ void k_build_xc(const float* __restrict__ img, const float* __restrict__ exem,
                           float* __restrict__ Xc) {
    size_t g = (size_t)blockIdx.x * blockDim.x + threadIdx.x;   // B*C*NPAD
    int n  = (int)(g % NPAD);
    int bc = (int)(g / NPAD);                                   // b*256 + c
    float v;
    if (n < HW_)        v = img[(size_t)bc * HW_ + n];
    else if (n == NEXE) v = exem[bc];
    else                v = 0.f;
    Xc[g] = v;
}

// ---------------------------------------------------------------------------
// helper: stage a 16-column tile of a [C_ x rowstride] matrix into LDS,
// transposed to At[j][k] (row stride ATP). 128 threads, float4 global reads.
// ---------------------------------------------------------------------------
static __device__ __forceinline__ void stage_at(float* At, const float* __restrict__ src,
                                                size_t rowstride, int col0, int tid) {
    for (int i = tid; i < (C_ * 16) / 4; i += 128) {
        const int k  = i >> 2;               // 0..255
        const int j4 = (i & 3) << 2;         // 0,4,8,12
        const float4 f = *(const float4*)&src[(size_t)k * rowstride + col0 + j4];
        At[(j4 + 0) * ATP + k] = f.x;
        At[(j4 + 1) * ATP + k] = f.y;
        At[(j4 + 2) * ATP + k] = f.z;
        At[(j4 + 3) * ATP + k] = f.w;
    }
}

// ---------------------------------------------------------------------------
// 3) U = X^T * S   (U[b,m,d] = sum_c img[b,c,m] * S[c,d])  -- fp32 WMMA
// ---------------------------------------------------------------------------
__global__ void k_gemm_u(const float* __restrict__ X, const float* __restrict__ S,
                         float* __restrict__ U) {
    __shared__ float At[16 * ATP];
    const int tid  = threadIdx.x;
    const int wv   = tid >> 5;
    const int lane = tid & 31, lo = lane & 15, hi = lane >> 4;
    const int blk = blockIdx.x;            // b(2) | mt(8) | dg(2)
    const int dg = blk & 3;
    const int mt = (blk >> 2) & 255;
    const int b  = blk >> 10;
    const int m0 = mt << 4;
    const int d0 = dg * 64 + wv * 16;

    const float* Xb = X + (size_t)b * C_ * HW_;
    stage_at(At, Xb, HW_, m0, tid);
    __syncthreads();

    v8f acc = {0.f,0.f,0.f,0.f,0.f,0.f,0.f,0.f};
    for (int k = 0; k < C_; k += 4) {
        v2f a, bb;
        const float2 av = *(const float2*)&At[lo * ATP + k + 2*hi];   // ds_load_b64
        a.x = av.x; a.y = av.y;
        bb.x = S[(k + 2*hi    ) * C_ + d0 + lo];
        bb.y = S[(k + 2*hi + 1) * C_ + d0 + lo];
        acc = wmma4(a, bb, acc);
    }
    float* Ub = U + (size_t)b * HW_ * C_;
    #pragma unroll
    for (int r = 0; r < 8; ++r)
        Ub[(size_t)(m0 + r + 8*hi) * C_ + d0 + lo] = acc[r];
}

// ---------------------------------------------------------------------------
// 4) V = Xc^T * Wv^T  (V[b,n,d] = sum_c Xc[b,c,n] * Wv[d,c]) -- fp32 WMMA
// ---------------------------------------------------------------------------
__global__ void k_gemm_v(const float* __restrict__ Xc, const float* __restrict__ Wv,
                         float* __restrict__ V) {
    __shared__ float At[16 * ATP];
    const int tid  = threadIdx.x;
    const int wv   = tid >> 5;
    const int lane = tid & 31, lo = lane & 15, hi = lane >> 4;
    const int blk = blockIdx.x;            // b * (257*4) + nt*4 + dg
    const int b   = blk / (NT_ * 4);
    const int rem = blk - b * (NT_ * 4);
    const int nt  = rem >> 2;
    const int dg  = rem & 3;
    const int n0 = nt << 4;
    const int d0 = dg * 64 + wv * 16;

    const float* Xb = Xc + (size_t)b * C_ * NPAD;
    stage_at(At, Xb, NPAD, n0, tid);
    __syncthreads();

    v8f acc = {0.f,0.f,0.f,0.f,0.f,0.f,0.f,0.f};
    for (int k = 0; k < C_; k += 4) {
        v2f a, bb;
        const float2 av = *(const float2*)&At[lo * ATP + k + 2*hi];   // ds_load_b64
        a.x = av.x; a.y = av.y;
        const float2 bv = *(const float2*)&Wv[(size_t)(d0 + lo) * C_ + k + 2*hi];
        bb.x = bv.x; bb.y = bv.y;
        acc = wmma4(a, bb, acc);
    }
    float* Vb = V + (size_t)b * NPAD * C_;
    #pragma unroll
    for (int r = 0; r < 8; ++r)
        Vb[(size_t)(n0 + r + 8*hi) * C_ + d0 + lo] = acc[r];
}

// ---------------------------------------------------------------------------
// 5) Fused flash attention: scores -> mask bias -> online softmax -> P*V.
//    Block = 4 waves on 4 consecutive m-tiles of one batch, marching over
//    n-tiles in lockstep. Shared B-tiles live in LDS:
//      XcS[k][j]  : 256 x 16 score B tile (natural layout)
//      Vs [nl][d] : 16 x 256 PV B tile, row stride VSP=264 (bank spread)
//    Staged per n-tile via TDM (tensor_load_to_lds + s_wait_tensorcnt) when
//    available, else cooperative float4 copies.
// ---------------------------------------------------------------------------
__global__ void __launch_bounds__(128, 1)
k_flash(const float* __restrict__ U, const float* __restrict__ Xc,
        const float* __restrict__ V, const float* __restrict__ mask,
        const float* __restrict__ img, float* __restrict__ out) {
    __shared__ float XcS[C_ * 16];         // 16 KB
    __shared__ float Vs [16 * VSP];        // 16.5 KB
    __shared__ float Plds[4][16 * 16];
    __shared__ float corr[4][16];
    __shared__ float lrow[4][16];

    const int tid  = threadIdx.x;
    const int wv   = tid >> 5;
    const int lane = tid & 31, lo = lane & 15, hi = lane >> 4;

    const int b  = blockIdx.x >> 6;                 // 4 batches
    const int mt = ((blockIdx.x & 63) << 2) | wv;   // 256 m-tiles
    const int m0 = mt << 4;

    const float* Ub = U  + (size_t)b * HW_ * C_;
    const float* Xb = Xc + (size_t)b * C_ * NPAD;
    const float* Vb = V  + (size_t)b * NPAD * C_;
    const float* mb = mask + (size_t)b * HW_;
    const float* Urow = Ub + (size_t)(m0 + lo) * C_;  // this lane's A-row

    const int   row  = lo;                 // softmax row owned by this lane (dup in halves)
    const float mrow = mb[m0 + row];

    v8f O[16];
    #pragma unroll
    for (int t = 0; t < 16; ++t) O[t] = (v8f){0.f,0.f,0.f,0.f,0.f,0.f,0.f,0.f};
    float m_run = -1.0e30f, l_run = 0.f;

    for (int nt = 0; nt < NT_; ++nt) {
        const int n0 = nt << 4;

        // ---- stage the shared B tiles for this n-tile ----
        __syncthreads();                       // previous tiles fully consumed
#ifdef TDM_OK
        if (wv == 0)                           // score B tile: 256 rows x 16 cols of Xc
            tdm_load_2d(Xb + n0, (unsigned)(size_t)&XcS[0],
                        (2u << 16),                       // data_size=4B
                        /*tx*/16, /*ty*/C_, /*d0*/NPAD, /*d1*/C_, /*s0*/NPAD);
        if (wv == 1)                           // PV B tile: 16 rows x 256 cols of V,
            tdm_load_2d(Vb + (size_t)n0 * C_, (unsigned)(size_t)&Vs[0],
                        (2u << 16) | (1u << 20) | (7u << 22) | (7u << 25),
                        /*tx*/C_, /*ty*/16, /*d0*/C_, /*d1*/NPAD, /*s0*/C_);
                        // pad: every 256 dwords insert 8 dwords -> row stride VSP
        __builtin_amdgcn_s_wait_tensorcnt(0);
        __syncthreads();                       // tiles visible to all waves
#else
        for (int i = tid; i < (C_ * 16) / 4; i += 128) {      // XcS[k][j]
            const int k = i >> 2, j4 = (i & 3) << 2;
            *(float4*)&XcS[k * 16 + j4] =
                *(const float4*)&Xb[(size_t)k * NPAD + n0 + j4];
        }
        for (int i = tid; i < (16 * C_) / 4; i += 128) {      // Vs[nl][d]
            const int nl = i >> 6, d4 = (i & 63) << 2;
            *(float4*)&Vs[nl * VSP + d4] =
                *(const float4*)&Vb[(size_t)(n0 + nl) * C_ + d4];
        }
        __syncthreads();
#endif

        if (nt + 1 < NT_) {                    // global_prefetch_b8 of next tiles
            __builtin_prefetch(Xb + (size_t)(2*hi) * NPAD + n0 + 16, 0, 1);
            __builtin_prefetch(Vb + (size_t)(n0 + 16 + hi) * C_, 0, 1);
        }

        // ---- scores tile S16x16 = U[m,:] . Xc[:,n]  (64 wmma, K=256) ----
        v8f sa = (v8f){0.f,0.f,0.f,0.f,0.f,0.f,0.f,0.f};
        for (int k = 0; k < C_; k += 4) {
            v2f a, bb;
            const float2 av = *(const float2*)(Urow + k + 2*hi);          // global b64
            a.x = av.x; a.y = av.y;
            bb.x = XcS[(k + 2*hi    ) * 16 + lo];                         // ds 2addr
            bb.y = XcS[(k + 2*hi + 1) * 16 + lo];
            sa = wmma4(a, bb, sa);
        }
        // spill C/D layout to LDS as P[m][j]
        #pragma unroll
        for (int r = 0; r < 8; ++r)
            Plds[wv][(r + 8*hi) * 16 + lo] = sa[r];

        // ---- per-row: bias, heatmap capture, online softmax update ----
        float svals[16];
        float tmax = -1.0e30f;
        #pragma unroll
        for (int j = 0; j < 16; ++j) {
            float s = Plds[wv][row * 16 + j];
            const int n = n0 + j;
            if (n == NEXE) {
                // pre-mask exemplar score == heatmap channel (ch 256)
                out[((size_t)(b * (C_ + 1) + C_)) * HW_ + m0 + row] = s;
            } else if (n > NEXE) {
                s = -2.0e9f;                               // padding column
            } else {
                const float mn = mb[n];
                if (mrow * mn != 1.f) s -= NEGB;           // mask bias
            }
            svals[j] = s;
            tmax = fmaxf(tmax, s);
        }
        const float nm = fmaxf(m_run, tmax);
        const float cf = __expf(m_run - nm);
        float ls = 0.f;
        #pragma unroll
        for (int j = 0; j < 16; ++j) {
            const float p = __expf(svals[j] - nm);
            Plds[wv][row * 16 + j] = p;                    // P tile for WMMA
            ls += p;
        }
        l_run = l_run * cf + ls;
        m_run = nm;
        corr[wv][row] = cf;

        // ---- rescale accumulators by per-row correction ----
        float cvec[8];
        #pragma unroll
        for (int r = 0; r < 8; ++r) cvec[r] = corr[wv][r + 8*hi];
        #pragma unroll
        for (int t = 0; t < 16; ++t)
            #pragma unroll
            for (int r = 0; r < 8; ++r) O[t][r] *= cvec[r];

        // ---- O += P(16x16) * V(16x256)  (64 wmma, K=16) ----
        v2f ap[4];
        #pragma unroll
        for (int kk = 0; kk < 4; ++kk) {
            ap[kk].x = Plds[wv][lo * 16 + kk*4 + 2*hi];
            ap[kk].y = Plds[wv][lo * 16 + kk*4 + 2*hi + 1];
        }
        #pragma unroll
        for (int t = 0; t < 16; ++t) {
            v8f acc = O[t];
            #pragma unroll
            for (int kk = 0; kk < 4; ++kk) {
                v2f bb;
                bb.x = Vs[(kk*4 + 2*hi    ) * VSP + t*16 + lo];
                bb.y = Vs[(kk*4 + 2*hi + 1) * VSP + t*16 + lo];
                acc = wmma4(ap[kk], bb, acc);
            }
            O[t] = acc;
        }
    }

    // ---- epilogue: normalize by l, residual add, write img_out (ch-major) ----
    lrow[wv][row] = l_run;
    float lv[8];
    #pragma unroll
    for (int r = 0; r < 8; ++r) lv[r] = 1.f / lrow[wv][r + 8*hi];
    #pragma unroll
    for (int t = 0; t < 16; ++t) {
        const int ch = t * 16 + lo;
        #pragma unroll
        for (int r = 0; r < 8; ++r) {
            const int m = m0 + r + 8*hi;
            out[((size_t)(b * (C_ + 1) + ch)) * HW_ + m] =
                O[t][r] * lv[r] + img[((size_t)(b * C_ + ch)) * HW_ + m];
        }
    }
}

// ---------------------------------------------------------------------------
// 6) NMS suppression map from heatmap channel of img_out
// ---------------------------------------------------------------------------
__global__ void k_nms(const float* __restrict__ out, float* __restrict__ sup) {
    int g = blockIdx.x * blockDim.x + threadIdx.x;      // B*HW
    int b = g >> 12, m = g & (HW_ - 1);
    int y = m >> 6, x = m & 63;
    const float* hmb = out + ((size_t)(b * (C_ + 1) + C_)) * HW_;
    float s = hmb[m];
    bool ismax = false;
    if (y >= 1 && y <= 62 && x >= 1 && x <= 62) {
        float up = hmb[m - W_], dn = hmb[m + W_];
        float lf = hmb[m - 1],  rt = hmb[m + 1];
        ismax = (s >= dn) && (s > up) && (s >= rt) && (s > lf);
    }
    sup[g] = ismax ? s : s - NEGB;
}

// ---------------------------------------------------------------------------
// 7) top-150 per batch (iterative argmax; ties -> lowest index, like lax.top_k)
// ---------------------------------------------------------------------------
__global__ void k_topk(float* __restrict__ sup, int* __restrict__ qids,
                       float* __restrict__ coords) {
    __shared__ float bv[256];
    __shared__ int   bi[256];
    const int b = blockIdx.x, t = threadIdx.x;
    float* s = sup + (size_t)b * HW_;
    for (int it = 0; it < Q_; ++it) {
        float best = -3.0e38f; int bidx = HW_;
        for (int i = t; i < HW_; i += 256) {
            float v = s[i];
            if (v > best) { best = v; bidx = i; }   // increasing i => first max kept
        }
        bv[t] = best; bi[t] = bidx;
        __syncthreads();
        for (int st = 128; st > 0; st >>= 1) {
            if (t < st) {
                if (bv[t + st] > bv[t] ||
                    (bv[t + st] == bv[t] && bi[t + st] < bi[t])) {
                    bv[t] = bv[t + st]; bi[t] = bi[t + st];
                }
            }
            __syncthreads();
        }
        if (t == 0) {
            int idx = bi[0];
            qids[b * Q_ + it] = idx;
            coords[(size_t)(b * Q_ + it) * 2 + 0] = (float)(idx >> 6);
            coords[(size_t)(b * Q_ + it) * 2 + 1] = (float)(idx & 63);
            s[idx] = -3.0e38f;
        }
        __syncthreads();
    }
}

// ---------------------------------------------------------------------------
// 8) gather queries + LayerNorm(257)
// ---------------------------------------------------------------------------
__global__ void k_gather_ln(const float* __restrict__ out, const int* __restrict__ qids,
                            const float* __restrict__ g, const float* __restrict__ be,
                            float* __restrict__ qn) {
    int gid = blockIdx.x * blockDim.x + threadIdx.x;    // B*Q
    if (gid >= B_ * Q_) return;
    int b = gid / Q_;
    int id = qids[gid];
    const float* src = out + (size_t)b * CP1 * HW_ + id;  // stride HW_ per channel
    float mu = 0.f;
    for (int ch = 0; ch < CP1; ++ch) mu += src[(size_t)ch * HW_];
    mu *= (1.0f / CP1);
    float var = 0.f;
    for (int ch = 0; ch < CP1; ++ch) {
        float d = src[(size_t)ch * HW_] - mu;
        var += d * d;
    }
    var *= (1.0f / CP1);
    float inv = rsqrtf(var + 1e-5f);
    float* q = qn + (size_t)gid * CP1;
    for (int ch = 0; ch < CP1; ++ch)
        q[ch] = (src[(size_t)ch * HW_] - mu) * inv * g[ch] + be[ch];
}

// ---------------------------------------------------------------------------
// 9) MLP: hid = relu(qn @ W1^T + b1) ;  q_out = hid @ W2^T + b2
// ---------------------------------------------------------------------------
__global__ void k_mlp1(const float* __restrict__ qn, const float* __restrict__ W1,
                       const float* __restrict__ b1, float* __restrict__ hid) {
    int gid = blockIdx.x * blockDim.x + threadIdx.x;    // B*Q*HID
    int h  = gid & (HID_ - 1);
    int bi = gid >> 10;
    const float* q = qn + (size_t)bi * CP1;
    const float* w = W1 + (size_t)h * CP1;
    float acc = b1[h];
    for (int j = 0; j < CP1; ++j) acc += q[j] * w[j];
    hid[gid] = fmaxf(acc, 0.f);
}

__global__ void k_mlp2(const float* __restrict__ hid, const float* __restrict__ W2,
                       const float* __restrict__ b2, float* __restrict__ qout) {
    int gid = blockIdx.x * blockDim.x + threadIdx.x;    // B*Q*CP1
    if (gid >= B_ * Q_ * CP1) return;
    int o  = gid % CP1;
    int bi = gid / CP1;
    const float* hp = hid + (size_t)bi * HID_;
    const float* w  = W2 + (size_t)o * HID_;
    float acc = b2[o];
    for (int h = 0; h < HID_; ++h) acc += hp[h] * w[h];
    qout[gid] = acc;
}

// ---------------------------------------------------------------------------
// Host-side orchestration
// ---------------------------------------------------------------------------
extern "C" void kernel_launch(void* const* d_in, const int* in_sizes, int n_in,
                              void* d_out, int out_size, void* d_ws, size_t ws_size,
                              hipStream_t stream) {
    (void)in_sizes; (void)n_in; (void)out_size; (void)ws_size;
    const float* img  = (const float*)d_in[0];   // [B,C,H,W]
    const float* exe  = (const float*)d_in[1];   // [B,C,7,7]
    const float* mask = (const float*)d_in[2];   // [B,H,W]
    const float* S    = (const float*)d_in[3];   // [C,C]
    const float* Wv   = (const float*)d_in[4];   // [C,C]
    const float* ln_g = (const float*)d_in[5];   // [257]
    const float* ln_b = (const float*)d_in[6];   // [257]
    const float* W1   = (const float*)d_in[7];   // [1024,257]
    const float* b1   = (const float*)d_in[8];   // [1024]
    const float* W2   = (const float*)d_in[9];   // [257,1024]
    const float* b2   = (const float*)d_in[10];  // [257]

    float* ws = (float*)d_ws;
    // workspace layout (float offsets)
    const size_t OFF_EXE  = 0;                                   //  1024
    const size_t OFF_XC   = OFF_EXE  + (size_t)B_ * C_;          //  B*C*NPAD
    const size_t OFF_U    = OFF_XC   + (size_t)B_ * C_ * NPAD;   //  B*HW*C
    const size_t OFF_V    = OFF_U    + (size_t)B_ * HW_ * C_;    //  B*NPAD*C
    const size_t OFF_SUP  = OFF_V    + (size_t)B_ * NPAD * C_;   //  B*HW
    const size_t OFF_QID  = OFF_SUP  + (size_t)B_ * HW_;         //  B*Q ints
    const size_t OFF_QN   = OFF_QID  + (size_t)B_ * Q_;          //  B*Q*257
    const size_t OFF_HID  = OFF_QN   + (size_t)B_ * Q_ * CP1;    //  B*Q*1024

    float* exem = ws + OFF_EXE;
    float* Xc   = ws + OFF_XC;
    float* U    = ws + OFF_U;
    float* V    = ws + OFF_V;
    float* sup  = ws + OFF_SUP;
    int*   qids = (int*)(ws + OFF_QID);
    float* qn   = ws + OFF_QN;
    float* hid  = ws + OFF_HID;

    float* out_img   = (float*)d_out;                                   // [B,257,HW]
    float* out_q     = out_img + (size_t)B_ * CP1 * HW_;                // [B,Q,257]
    float* out_coord = out_q   + (size_t)B_ * Q_ * CP1;                 // [B,Q,2]

    k_exe_mean <<<4,     256, 0, stream>>>(exe, exem);
    k_build_xc <<<16448, 256, 0, stream>>>(img, exem, Xc);
    k_gemm_u   <<<4096,  128, 0, stream>>>(img, S, U);
    k_gemm_v   <<<4112,  128, 0, stream>>>(Xc, Wv, V);
    k_flash    <<<256,   128, 0, stream>>>(U, Xc, V, mask, img, out_img);
    k_nms      <<<64,    256, 0, stream>>>(out_img, sup);
    k_topk     <<<4,     256, 0, stream>>>(sup, qids, out_coord);
    k_gather_ln<<<3,     256, 0, stream>>>(out_img, qids, ln_g, ln_b, qn);
    k_mlp1     <<<2400,  256, 0, stream>>>(qn, W1, b1, hid);
    k_mlp2     <<<603,   256, 0, stream>>>(hid, W2, b2, out_q);
}